// CharWordBiLSTMCRF_16853451670189
// MI455X (gfx1250) — compile-verified
//
#include <hip/hip_runtime.h>
#include <math.h>

// ---------------- CDNA5 WMMA fp32 16x16x4 ----------------
typedef float v2f __attribute__((ext_vector_type(2)));
typedef float v8f __attribute__((ext_vector_type(8)));

#define WMMA_F32(a, b, c) \
  __builtin_amdgcn_wmma_f32_16x16x4_f32(false, (a), false, (b), (short)0, (c), false, false)

static __device__ __forceinline__ float sigmoidf_(float x) {
  return 1.0f / (1.0f + __expf(-x));
}

// Problem constants: B=32, T=512, E=128, D=256, H=128 (per dir), 4H=512, K_tags=15

// ---------------------------------------------------------------------------
// 1) Embedding gather + fuse
// ---------------------------------------------------------------------------
__global__ void embed_fuse(const int* __restrict__ cids, const int* __restrict__ wids,
                           const float* __restrict__ cemb, const float* __restrict__ wemb,
                           float* __restrict__ xs) {
  int g = blockIdx.x * blockDim.x + threadIdx.x;
  if (g >= 512 * 32 * 64) return;
  int c4  = g & 63;        // which float4 of the 256-wide row
  int row = g >> 6;        // t*32 + b
  int t = row >> 5, b = row & 31;
  float4 v;
  int dstoff;
  if (c4 < 32) {
    int id = cids[b * 512 + t];
    v = ((const float4*)(cemb + (size_t)id * 128))[c4];
    dstoff = row * 256 + c4 * 4;
  } else {
    int id = wids[b * 512 + t];
    v = ((const float4*)(wemb + (size_t)id * 128))[c4 - 32];
    dstoff = row * 256 + 128 + (c4 - 32) * 4;
  }
  *(float4*)(xs + dstoff) = v;
}

// ---------------------------------------------------------------------------
// 2) LDS-tiled fp32 WMMA GEMM:  C[M,N] = X[M,K] * W[N,K]^T + b0[N] + b1[N]
//    Block tile 64(M) x 128(N), K-slab 16. 256 threads = 8 waves.
// ---------------------------------------------------------------------------
__global__ __launch_bounds__(256) void gemm_bias_wmma(
    const float* __restrict__ X, const float* __restrict__ W,
    const float* __restrict__ b0, const float* __restrict__ b1,
    float* __restrict__ C, int M, int N, int K) {
  __shared__ float As[64 * 16];
  __shared__ float Bs[128 * 16];
  const int tid = threadIdx.x;
  const int lid = tid & 31;
  const int wid = tid >> 5;
  const int m0 = blockIdx.y * 64;
  const int n0 = blockIdx.x * 128;
  const int wm = wid & 3;   // 0..3
  const int wn = wid >> 2;  // 0..1
  const int nlo = lid & 15;
  const int hsel = lid >> 4;  // 0 or 1

  v8f acc[4];
  for (int s = 0; s < 4; ++s)
    for (int r = 0; r < 8; ++r) acc[s][r] = 0.0f;

  for (int k0 = 0; k0 < K; k0 += 16) {
    {  // stage A slab: 64x16 -> 256 float4, one per thread
      int r = tid >> 2, c = (tid & 3) * 4;
      float4 v = *(const float4*)(X + (size_t)(m0 + r) * K + k0 + c);
      *(float4*)&As[r * 16 + c] = v;
    }
    for (int j = 0; j < 2; ++j) {  // stage B slab: 128x16 -> 512 float4
      int idx = tid + j * 256;
      int r = idx >> 2, c = (idx & 3) * 4;
      float4 v = *(const float4*)(W + (size_t)(n0 + r) * K + k0 + c);
      *(float4*)&Bs[r * 16 + c] = v;
    }
    __syncthreads();
    const int arow = wm * 16 + nlo;
    for (int kk = 0; kk < 16; kk += 4) {
      int ko = kk + hsel * 2;
      v2f a;
      a.x = As[arow * 16 + ko];
      a.y = As[arow * 16 + ko + 1];
      for (int s = 0; s < 4; ++s) {
        int bcol = wn * 64 + s * 16 + nlo;
        v2f bfr;
        bfr.x = Bs[bcol * 16 + ko];
        bfr.y = Bs[bcol * 16 + ko + 1];
        acc[s] = WMMA_F32(a, bfr, acc[s]);
      }
    }
    __syncthreads();
  }
  // epilogue with fused dual bias
  for (int s = 0; s < 4; ++s) {
    int col = n0 + wn * 64 + s * 16 + nlo;
    float bias = b0[col] + b1[col];
    for (int r = 0; r < 8; ++r) {
      int row = m0 + wm * 16 + r + 8 * hsel;
      C[(size_t)row * N + col] = acc[s][r] + bias;
    }
  }
}

// ---------------------------------------------------------------------------
// 3) Sequential (bi)LSTM, one block per direction. 512 threads = 16 waves.
//    Gate-aligned tile ownership: wave (mi, ti) computes the four 16x16
//    output tiles {ti, ti+8, ti+16, ti+24} of g = h @ Whh^T, i.e. all four
//    gates (i,f,g,o) of columns j = ti*16..ti*16+15 for rows mi*16..mi*16+15.
//    After the WMMA k-loop each lane holds gi/gf/gc/go of its (b, j) elements
//    in acc[gate][r] -> gate math fully in registers, c persistent in regs.
//    LDS: only h (32x128 f32 = 16KB). Whh streamed from L2 (256KB/dir).
// ---------------------------------------------------------------------------
__global__ __launch_bounds__(512) void lstm_seq(
    const float* __restrict__ pre,   // (T*32) x 1024 row-major; this dir at col dir*512
    const float* __restrict__ Whh,   // (2, 512, 128) row-major
    float* __restrict__ hs,          // (T*32) x 256; this dir at col dir*128
    int T) {
  extern __shared__ float lds[];
  float* h_lds = lds;  // 32*128
  const int dir = blockIdx.x;
  const float* Wd = Whh + (size_t)dir * 512 * 128;
  const int tid = threadIdx.x, lid = tid & 31, wid = tid >> 5;
  const int nlo = lid & 15;
  const int hsel = lid >> 4;
  const int mi = wid >> 3;  // 0..1  (m-tile)
  const int ti = wid & 7;   // 0..7  (column group within a gate block)
  const int jcol = ti * 16 + nlo;            // 0..127 column within gate
  const int arowbase = (mi * 16 + nlo) * 128;

  for (int i = tid; i < 32 * 128; i += 512) h_lds[i] = 0.0f;
  float c_reg[8];
  for (int r = 0; r < 8; ++r) c_reg[r] = 0.0f;
  __syncthreads();

  for (int step = 0; step < T; ++step) {
    const int t = dir ? (T - 1 - step) : step;

    // ---- gate GEMM: four 16x16 tiles (one per gate), K=128, fp32 WMMA ----
    v8f acc[4];
    for (int g = 0; g < 4; ++g)
      for (int r = 0; r < 8; ++r) acc[g][r] = 0.0f;
#pragma unroll 4
    for (int kk = 0; kk < 128; kk += 4) {
      int ko = kk + hsel * 2;
      v2f a;
      a.x = h_lds[arowbase + ko];
      a.y = h_lds[arowbase + ko + 1];
#pragma unroll
      for (int g = 0; g < 4; ++g) {
        int col = g * 128 + jcol;  // Whh row (output unit)
        v2f bfr;
        bfr.x = Wd[(size_t)col * 128 + ko];
        bfr.y = Wd[(size_t)col * 128 + ko + 1];
        acc[g] = WMMA_F32(a, bfr, acc[g]);
      }
    }
    __syncthreads();  // all waves done reading h_lds for this step

    // ---- fused gates entirely in registers ----
    const size_t prow = (size_t)t * 32 * 1024 + (size_t)dir * 512;
#pragma unroll
    for (int r = 0; r < 8; ++r) {
      int b = mi * 16 + r + 8 * hsel;  // batch row this lane owns
      size_t pb = prow + (size_t)b * 1024 + jcol;
      float gi = acc[0][r] + pre[pb];
      float gf = acc[1][r] + pre[pb + 128];
      float gc = acc[2][r] + pre[pb + 256];
      float go = acc[3][r] + pre[pb + 384];
      float cv = sigmoidf_(gf) * c_reg[r] + sigmoidf_(gi) * tanhf(gc);
      c_reg[r] = cv;
      float hv = sigmoidf_(go) * tanhf(cv);
      h_lds[b * 128 + jcol] = hv;
      hs[((size_t)t * 32 + b) * 256 + dir * 128 + jcol] = hv;
    }
    __syncthreads();  // h_lds ready for next step
  }
}

// ---------------------------------------------------------------------------
// 4) Emission FC: em[(t*32+b)*15 + k] = xs . fc_w[k] + fc_b[k]
// ---------------------------------------------------------------------------
__global__ void fc_kernel(const float* __restrict__ X, const float* __restrict__ W,
                          const float* __restrict__ bias, float* __restrict__ em) {
  int g = blockIdx.x * blockDim.x + threadIdx.x;
  if (g >= 16384 * 15) return;
  int row = g / 15, k = g % 15;
  const float* x = X + (size_t)row * 256;
  const float* w = W + (size_t)k * 256;
  float s = bias[k];
  for (int i = 0; i < 256; ++i) s += x[i] * w[i];
  em[g] = s;
}

// ---------------------------------------------------------------------------
// 5) CRF NLL: forward algorithm + gold path score, single block.
// ---------------------------------------------------------------------------
__global__ __launch_bounds__(512) void crf_kernel(
    const float* __restrict__ em, const int* __restrict__ tags,
    const unsigned char* __restrict__ mask, const float* __restrict__ trans,
    const float* __restrict__ startv, const float* __restrict__ endv,
    float* __restrict__ out) {
  __shared__ float s_tr[15 * 15];
  __shared__ float s_st[15], s_en[15];
  __shared__ float alpha[32 * 16], alpha2[32 * 16];
  __shared__ float s_red[32];
  int tid = threadIdx.x;
  if (tid < 225) s_tr[tid] = trans[tid];
  if (tid < 15) { s_st[tid] = startv[tid]; s_en[tid] = endv[tid]; }
  if (tid < 480) {
    int b = tid / 15, k = tid % 15;
    alpha[b * 16 + k] = startv[k] + em[b * 15 + k];
  }
  __syncthreads();
  for (int t = 1; t < 512; ++t) {
    if (tid < 480) {
      int b = tid / 15, kp = tid % 15;
      float mmax = -1e30f;
      for (int k = 0; k < 15; ++k)
        mmax = fmaxf(mmax, alpha[b * 16 + k] + s_tr[k * 15 + kp]);
      float s = 0.0f;
      for (int k = 0; k < 15; ++k)
        s += __expf(alpha[b * 16 + k] + s_tr[k * 15 + kp] - mmax);
      float nxt = mmax + __logf(s) + em[((size_t)t * 32 + b) * 15 + kp];
      float mt = (float)mask[b * 512 + t];
      alpha2[b * 16 + kp] = (mt > 0.0f) ? nxt : alpha[b * 16 + kp];
    }
    __syncthreads();
    if (tid < 480) {
      int b = tid / 15, kp = tid % 15;
      alpha[b * 16 + kp] = alpha2[b * 16 + kp];
    }
    __syncthreads();
  }
  if (tid < 32) {
    int b = tid;
    int tg0 = tags[b * 512];
    float num = s_st[tg0] + em[b * 15 + tg0];
    int send = 0;
    for (int t = 0; t < 512; ++t) send += (int)mask[b * 512 + t];
    send -= 1;
    for (int t = 1; t < 512; ++t) {
      int tg = tags[b * 512 + t], tp = tags[b * 512 + t - 1];
      float mt = (float)mask[b * 512 + t];
      num += (s_tr[tp * 15 + tg] + em[((size_t)t * 32 + b) * 15 + tg]) * mt;
    }
    num += s_en[tags[b * 512 + send]];
    float mmax = -1e30f;
    for (int k = 0; k < 15; ++k) mmax = fmaxf(mmax, alpha[b * 16 + k] + s_en[k]);
    float s = 0.0f;
    for (int k = 0; k < 15; ++k) s += __expf(alpha[b * 16 + k] + s_en[k] - mmax);
    float den = mmax + __logf(s);
    s_red[b] = den - num;
  }
  __syncthreads();
  if (tid == 0) {
    float tot = 0.0f;
    for (int b = 0; b < 32; ++b) tot += s_red[b];
    out[0] = tot;
  }
}

// ---------------------------------------------------------------------------
extern "C" void kernel_launch(void* const* d_in, const int* in_sizes, int n_in,
                              void* d_out, int out_size, void* d_ws, size_t ws_size,
                              hipStream_t stream) {
  const int* char_ids = (const int*)d_in[0];
  const int* word_ids = (const int*)d_in[1];
  const int* tags = (const int*)d_in[2];
  const unsigned char* mask = (const unsigned char*)d_in[3];
  const float* char_emb = (const float*)d_in[4];
  const float* word_emb = (const float*)d_in[5];
  const float* wih = (const float*)d_in[6];   // (2,2,512,256)
  const float* whh = (const float*)d_in[7];   // (2,2,512,128)
  const float* bih = (const float*)d_in[8];   // (2,2,512)
  const float* bhh = (const float*)d_in[9];
  const float* fc_w = (const float*)d_in[10];
  const float* fc_b = (const float*)d_in[11];
  const float* trans = (const float*)d_in[12];
  const float* start_t = (const float*)d_in[13];
  const float* end_t = (const float*)d_in[14];
  float* out = (float*)d_out;

  float* ws = (float*)d_ws;
  float* xs0 = ws;                             // 16384*256
  float* xs1 = xs0 + (size_t)16384 * 256;      // 16384*256
  float* pre = xs1 + (size_t)16384 * 256;      // 16384*1024
  float* em = pre + (size_t)16384 * 1024;      // 16384*15

  const int M = 16384, Npre = 1024, K = 256;
  dim3 ggrid(Npre / 128, M / 64);
  const size_t lstm_lds = (size_t)(32 * 128) * sizeof(float);  // 16 KB

  // layer 0
  embed_fuse<<<(512 * 32 * 64 + 255) / 256, 256, 0, stream>>>(
      char_ids, word_ids, char_emb, word_emb, xs0);
  gemm_bias_wmma<<<ggrid, 256, 0, stream>>>(xs0, wih, bih, bhh, pre, M, Npre, K);
  lstm_seq<<<2, 512, lstm_lds, stream>>>(pre, whh, xs1, 512);
  // layer 1 (input GEMM on layer-0 output; weights at layer offset)
  gemm_bias_wmma<<<ggrid, 256, 0, stream>>>(xs1, wih + (size_t)2 * 512 * 256,
                                            bih + 1024, bhh + 1024, pre, M, Npre, K);
  lstm_seq<<<2, 512, lstm_lds, stream>>>(pre, whh + (size_t)2 * 512 * 128, xs0, 512);
  // emissions + CRF
  fc_kernel<<<(16384 * 15 + 255) / 256, 256, 0, stream>>>(xs0, fc_w, fc_b, em);
  crf_kernel<<<1, 512, 0, stream>>>(em, tags, mask, trans, start_t, end_t, out);
}